// YOLOv1Loss_79259326480482
// MI455X (gfx1250) — compile-verified
//
#include <hip/hip_runtime.h>
#include <hip/hip_bf16.h>

typedef __attribute__((ext_vector_type(2))) float v2f;
typedef __attribute__((ext_vector_type(8))) float v8f;

#define S7 7
#define NCELLS (8192 * 49)
#define THREADS 256
#define BLOCKS 224
#define CPT 7   // cells per thread: 224*256*7 == 401408 exactly

__device__ __forceinline__ float clip01(float v) {
    return fminf(fmaxf(v, 0.0f), 1.0f);
}

__device__ __forceinline__ float cell_loss(const float* __restrict__ xc,
                                           const float* __restrict__ yc,
                                           float gi, float gj) {
    // 30 floats each, base is 8-byte aligned (cell*120 bytes) -> b64 loads
    float xv[30], yv[30];
    const float2* x2 = reinterpret_cast<const float2*>(xc);
    const float2* y2 = reinterpret_cast<const float2*>(yc);
#pragma unroll
    for (int k = 0; k < 15; ++k) {
        float2 a = x2[k]; xv[2 * k] = a.x; xv[2 * k + 1] = a.y;
        float2 b = y2[k]; yv[2 * k] = b.x; yv[2 * k + 1] = b.y;
    }

    const float obj   = (yv[4] > 0.0f)  ? 1.0f : 0.0f;
    const float noobj = (yv[4] == 0.0f) ? 1.0f : 0.0f;
    const float invS = 1.0f / 7.0f;

    // target box 0 in grid-normalized coords, clipped xyxy
    const float t_ux = (yv[0] + gi) * invS;
    const float t_uy = (yv[1] + gj) * invS;
    const float tw0 = yv[2], th0 = yv[3];
    const float t_x0 = clip01(t_ux - 0.5f * tw0);
    const float t_y0 = clip01(t_uy - 0.5f * th0);
    const float t_x1 = clip01(t_ux + 0.5f * tw0);
    const float t_y1 = clip01(t_uy + 0.5f * th0);
    const float area_t = (t_x1 - t_x0) * (t_y1 - t_y0);

    float iou[2], ux[2], uy[2];
#pragma unroll
    for (int b = 0; b < 2; ++b) {
        const float px = xv[5 * b + 0], py = xv[5 * b + 1];
        const float pw = xv[5 * b + 2], ph = xv[5 * b + 3];
        ux[b] = (px + gi) * invS;
        uy[b] = (py + gj) * invS;
        const float p_x0 = clip01(ux[b] - 0.5f * pw);
        const float p_y0 = clip01(uy[b] - 0.5f * ph);
        const float p_x1 = clip01(ux[b] + 0.5f * pw);
        const float p_y1 = clip01(uy[b] + 0.5f * ph);
        const float lt_x = fmaxf(p_x0, t_x0), lt_y = fmaxf(p_y0, t_y0);
        const float rb_x = fminf(p_x1, t_x1), rb_y = fminf(p_y1, t_y1);
        const float iw = fmaxf(rb_x - lt_x, 0.0f);
        const float ih = fmaxf(rb_y - lt_y, 0.0f);
        const float inter = iw * ih;
        const float area_p = (p_x1 - p_x0) * (p_y1 - p_y0);
        iou[b] = inter / (area_p + area_t - inter);
    }

    // argmax (first occurrence on tie) + max
    const int best = (iou[1] > iou[0]) ? 1 : 0;
    const float max_iou = fmaxf(iou[0], iou[1]);

    // xy loss: box0 target is grid-normalized; box1 target is raw y[5],y[6]
    const float txu = (best == 0) ? t_ux : yv[5];
    const float tyu = (best == 0) ? t_uy : yv[6];
    const float dx = ux[best] - txu;
    const float dy = uy[best] - tyu;
    const float lxy = dx * dx + dy * dy;

    // wh loss (sqrt space)
    const float dw = sqrtf(xv[5 * best + 2]) - sqrtf(yv[5 * best + 2]);
    const float dh = sqrtf(xv[5 * best + 3]) - sqrtf(yv[5 * best + 3]);
    const float lwh = dw * dw + dh * dh;

    // objectness
    const float dob = xv[5 * best + 4] - max_iou;
    const float lobj = dob * dob;

    // class SSE (channels 10..29)
    float lcls = 0.0f;
#pragma unroll
    for (int c = 10; c < 30; ++c) {
        const float d = xv[c] - yv[c];
        lcls += d * d;
    }

    // no-object confidence (channels 4, 9)
    const float d4 = xv[4] - yv[4];
    const float d9 = xv[9] - yv[9];
    const float lno = d4 * d4 + d9 * d9;

    return obj * (5.0f * (lxy + lwh) + lobj + lcls) + 0.5f * noobj * lno;
}

__global__ void __launch_bounds__(THREADS)
yolo_loss_partial(const float* __restrict__ x, const float* __restrict__ y,
                  float* __restrict__ ws) {
    __shared__ float sdata[THREADS];
    const int tid = threadIdx.x;
    const int g = blockIdx.x * THREADS + tid;

    float acc = 0.0f;
#pragma unroll
    for (int k = 0; k < CPT; ++k) {
        const int cell = g + k * (BLOCKS * THREADS);
        const int rem = cell % 49;
        const float gi = (float)(rem / 7);
        const float gj = (float)(rem % 7);
        acc += cell_loss(x + (size_t)cell * 30, y + (size_t)cell * 30, gi, gj);
    }
    sdata[tid] = acc;
    __syncthreads();

    // Block reduction on the matrix pipe: D = ones(16x4) * B(4x16) + C.
    // Each WMMA folds 64 partials into 16 column sums; 4 accumulating WMMAs
    // cover all 256, then a 16-lane xor-shuffle collapse finishes.
    if (tid < 32) {                 // whole wave 0 => EXEC all ones inside
        const int lane = tid;
        v2f a; a.x = 1.0f; a.y = 1.0f;     // A = all-ones 16x4
        v8f c = {};
#pragma unroll
        for (int chunk = 0; chunk < 4; ++chunk) {
            // B layout (4x16 f32): lane l, vgpr j -> element (k=(l>>4)*2+j, n=l&15)
            const int base = chunk * 64 + ((lane >> 4) * 2) * 16 + (lane & 15);
            v2f b;
            b.x = sdata[base];
            b.y = sdata[base + 16];
            c = __builtin_amdgcn_wmma_f32_16x16x4_f32(
                    false, a, false, b, (short)0, c, false, false);
        }
        // c[0] holds colsum_n for n = lane%16 (rows identical since A==1)
        float v = c[0];
        v += __shfl_xor(v, 1, 32);
        v += __shfl_xor(v, 2, 32);
        v += __shfl_xor(v, 4, 32);
        v += __shfl_xor(v, 8, 32);
        if (lane == 0) ws[blockIdx.x] = v;
    }
}

__global__ void __launch_bounds__(THREADS)
yolo_loss_final(const float* __restrict__ ws, float* __restrict__ out) {
    __shared__ float s[THREADS];
    const int tid = threadIdx.x;
    s[tid] = (tid < BLOCKS) ? ws[tid] : 0.0f;
    __syncthreads();
#pragma unroll
    for (int off = THREADS / 2; off > 0; off >>= 1) {
        if (tid < off) s[tid] += s[tid + off];
        __syncthreads();
    }
    if (tid == 0) out[0] = s[0] * (1.0f / 8192.0f);
}

extern "C" void kernel_launch(void* const* d_in, const int* in_sizes, int n_in,
                              void* d_out, int out_size, void* d_ws, size_t ws_size,
                              hipStream_t stream) {
    (void)in_sizes; (void)n_in; (void)out_size; (void)ws_size;
    const float* x = (const float*)d_in[0];
    const float* y = (const float*)d_in[1];
    float* out = (float*)d_out;
    float* ws  = (float*)d_ws;   // 224 floats used

    yolo_loss_partial<<<BLOCKS, THREADS, 0, stream>>>(x, y, ws);
    yolo_loss_final<<<1, THREADS, 0, stream>>>(ws, out);
}